// KernelScoreLoss_56899726737831
// MI455X (gfx1250) — compile-verified
//
#include <hip/hip_runtime.h>
#include <math.h>

typedef __attribute__((ext_vector_type(2))) float v2f;
typedef __attribute__((ext_vector_type(8))) float v8f;

#define M      64
#define KTOT   (4096 * 128)     // S*D = 524288
#define BK     128              // K-chunk per block iteration
#define STRIDE 132              // padded LDS row stride (words), 132%64=4 -> conflict-free
#define NBLK   256
#define NCHUNK (KTOT / BK)      // 4096

#define GAMMA  1.0f
#define LAMBDA 0.5f

// ws layout (floats): [0 .. 4095] gram(64x64), [4096 .. 4159] xt, [4160] tt
#define WS_XT (M * M)
#define WS_TT (M * M + M)

__global__ __launch_bounds__(256)
void gram_fused_kernel(const float* __restrict__ X,
                       const float* __restrict__ t,
                       float* __restrict__ ws)
{
    __shared__ float lds_x[M * STRIDE];
    __shared__ float lds_red[M + 1];   // xt[64] + tt

    const int tid  = threadIdx.x;
    const int lane = tid & 31;
    const int wave = tid >> 5;

    // tile ownership: 16 tiles (4x4) of 16x16, 2 per wave sharing one A tile
    const int ti  = wave & 3;            // A row-tile
    const int tj0 = (wave >> 2) * 2;     // two B col-tiles: tj0, tj0+1

    v8f acc0 = {};
    v8f acc1 = {};
    float xt_part[8];
#pragma unroll
    for (int r = 0; r < 8; ++r) xt_part[r] = 0.0f;
    float tt_part = 0.0f;

    const int col4 = tid & 31;   // this thread's float4 column within the chunk
    const int row0 = tid >> 5;   // covers rows row0 + 8*r

    for (int chunk = blockIdx.x; chunk < NCHUNK; chunk += NBLK) {
        const long kbase = (long)chunk * BK;

        // t columns for this thread (fixed across its 8 rows)
        const float4 tv = *(const float4*)(t + kbase + col4 * 4);
        if (wave == 0) {  // each column covered exactly once by wave 0
            tt_part += tv.x * tv.x + tv.y * tv.y + tv.z * tv.z + tv.w * tv.w;
        }

        // stage X slab into LDS + accumulate X·t partials
#pragma unroll
        for (int r = 0; r < 8; ++r) {
            const int row = row0 + 8 * r;
            const float4 xv =
                *(const float4*)(X + (long)row * KTOT + kbase + col4 * 4);
            xt_part[r] += xv.x * tv.x + xv.y * tv.y + xv.z * tv.z + xv.w * tv.w;
            float* dst = &lds_x[row * STRIDE + col4 * 4];
            dst[0] = xv.x; dst[1] = xv.y; dst[2] = xv.z; dst[3] = xv.w;
        }

        // prefetch next chunk while we compute (global_prefetch_b8)
        if (chunk + NBLK < NCHUNK) {
            const long knext = kbase + (long)NBLK * BK;
            __builtin_prefetch(t + knext + col4 * 4, 0, 1);
#pragma unroll
            for (int r = 0; r < 8; ++r) {
                const int row = row0 + 8 * r;
                __builtin_prefetch(X + (long)row * KTOT + knext + col4 * 4, 0, 1);
            }
        }

        __syncthreads();

        // WMMA accumulation over this chunk.
        // A 16x4 f32 layout: lanes 0-15 hold K=kk,kk+1; lanes 16-31 hold K=kk+2,kk+3 (M=lane&15).
        // B 4x16 layout mirrors it with N=lane&15.
        const int l     = lane & 15;
        const int khalf = (lane >> 4) * 2;
        const float* arow  = &lds_x[(ti * 16 + l) * STRIDE + khalf];
        const float* brow0 = &lds_x[(tj0 * 16 + l) * STRIDE + khalf];
        const float* brow1 = &lds_x[((tj0 + 1) * 16 + l) * STRIDE + khalf];
#pragma unroll
        for (int kk = 0; kk < BK; kk += 4) {
            const v2f a  = *(const v2f*)(arow + kk);
            const v2f b0 = *(const v2f*)(brow0 + kk);
            const v2f b1 = *(const v2f*)(brow1 + kk);
            acc0 = __builtin_amdgcn_wmma_f32_16x16x4_f32(
                false, a, false, b0, (short)0, acc0, false, false);
            acc1 = __builtin_amdgcn_wmma_f32_16x16x4_f32(
                false, a, false, b1, (short)0, acc1, false, false);
        }

        __syncthreads();
    }

    // merge gram partials: C/D layout VGPR v -> M = v (lanes 0-15) / v+8 (lanes 16-31), N = lane&15
    {
        const int l   = lane & 15;
        const int mhi = (lane >> 4) * 8;
#pragma unroll
        for (int v = 0; v < 8; ++v) {
            const int row = ti * 16 + mhi + v;
            atomicAdd(&ws[row * M + tj0 * 16 + l], acc0[v]);
            atomicAdd(&ws[row * M + (tj0 + 1) * 16 + l], acc1[v]);
        }
    }

    // block-reduce xt / tt in LDS, then one global atomic per element
    if (tid < M + 1) lds_red[tid] = 0.0f;
    __syncthreads();
#pragma unroll
    for (int r = 0; r < 8; ++r)
        atomicAdd(&lds_red[row0 + 8 * r], xt_part[r]);
    if (wave == 0)
        atomicAdd(&lds_red[M], tt_part);
    __syncthreads();
    if (tid < M)  atomicAdd(&ws[WS_XT + tid], lds_red[tid]);
    if (tid == M) atomicAdd(&ws[WS_TT], lds_red[M]);
}

__global__ __launch_bounds__(64)
void finalize_kernel(const float* __restrict__ ws, float* __restrict__ out)
{
    __shared__ float sq[M];
    __shared__ float rc[M];
    __shared__ float rt[M];

    const int i = threadIdx.x;  // 64 threads, one row each
    const float sqi = ws[i * M + i];
    sq[i] = sqi;
    __syncthreads();

    float cross = 0.0f;
    for (int j = 0; j < M; ++j) {
        if (j == i) continue;
        float d2 = sqi + sq[j] - 2.0f * ws[i * M + j];
        d2 = fmaxf(d2, 0.0f);
        cross += expf(-GAMMA * d2);
    }

    const float tt  = ws[WS_TT];
    const float dt2 = sqi - 2.0f * ws[WS_XT + i] + tt;
    const float tgt = expf(-GAMMA * dt2);

    rc[i] = cross;
    rt[i] = tgt;
    __syncthreads();

    if (i == 0) {
        float cs = 0.0f, ts = 0.0f;
        for (int j = 0; j < M; ++j) { cs += rc[j]; ts += rt[j]; }
        const float cross_term  = (LAMBDA * 0.5f) * cs / (float)(M * (M - 1));
        const float target_term = ts / (float)M;
        float score = cross_term - target_term;
        score = fminf(fmaxf(score, -10.0f), 10.0f);
        out[0] = score;
    }
}

extern "C" void kernel_launch(void* const* d_in, const int* in_sizes, int n_in,
                              void* d_out, int out_size, void* d_ws, size_t ws_size,
                              hipStream_t stream)
{
    const float* X = (const float*)d_in[0];   // [64, 4096, 128] f32
    const float* t = (const float*)d_in[1];   // [4096, 128] f32
    // d_in[2] (time_points) cancels in the kernel math, as in the reference.
    float* ws = (float*)d_ws;

    hipMemsetAsync(d_ws, 0, (size_t)(M * M + M + 1) * sizeof(float), stream);
    gram_fused_kernel<<<NBLK, 256, 0, stream>>>(X, t, ws);
    finalize_kernel<<<1, 64, 0, stream>>>(ws, (float*)d_out);
}